// GNN_2946347566021
// MI455X (gfx1250) — compile-verified
//
#include <hip/hip_runtime.h>
#include <hip/hip_bf16.h>

typedef __attribute__((ext_vector_type(2))) float v2f;
typedef __attribute__((ext_vector_type(8))) float v8f;

// ---------------- utility kernels ----------------
__global__ void k_fill(float* __restrict__ p, float v, int n) {
    int i = blockIdx.x * blockDim.x + threadIdx.x;
    if (i < n) p[i] = v;
}

__global__ void k_deg(const int* __restrict__ dst, float* __restrict__ deg, int E) {
    int e = blockIdx.x * blockDim.x + threadIdx.x;
    if (e < E) atomicAdd(&deg[dst[e]], 1.0f);
}

__global__ void k_rsqrt(float* __restrict__ d, int n) {
    int i = blockIdx.x * blockDim.x + threadIdx.x;
    if (i < n) {
        float x = d[i];                      // deg >= 1 always (self loops)
        float r = rsqrtf(x);
        r = r * (1.5f - 0.5f * x * r * r);   // one Newton step for accuracy
        d[i] = r;
    }
}

__global__ void k_norm(const int* __restrict__ src, const int* __restrict__ dst,
                       const float* __restrict__ dinv, float* __restrict__ nrm, int E) {
    int e = blockIdx.x * blockDim.x + threadIdx.x;
    if (e < E) nrm[e] = dinv[src[e]] * dinv[dst[e]];
}

__global__ void k_cnt(const int* __restrict__ batch, float* __restrict__ cnt, int n) {
    int i = blockIdx.x * blockDim.x + threadIdx.x;
    if (i < n) atomicAdd(&cnt[batch[i]], 1.0f);
}

// ---------------- WMMA GEMM:  H[N x 64] = act(X)[N x KDIM] @ W[KDIM x 64] ----------------
// One wave computes a 16-row x 64-col tile using V_WMMA_F32_16X16X4_F32.
// W is staged once per block into LDS (stride 65 to dodge the 2-way bank conflict
// between lane halves), zero-padded so the inner loop has no K guards. OOB A rows
// are clamped (their outputs are never stored), and the one padded A element in
// the K=15 case is fetched from a clamped address and selected to zero -> the
// entire hot loop is branch-free with EXEC all-1s.
template <int KDIM, bool RELU>
__global__ void k_gemm(const float* __restrict__ X, const float* __restrict__ W,
                       float* __restrict__ H, int nrows) {
    constexpr int KSTEPS = (KDIM + 3) / 4;   // 4 (K=15 padded to 16) or 16 (K=64)
    constexpr int KPAD   = KSTEPS * 4;
    __shared__ float wlds[KPAD * 65];

    // cooperative stage of W -> LDS, zero-padded
    for (int idx = threadIdx.x; idx < KPAD * 64; idx += blockDim.x) {
        const int k = idx >> 6, n = idx & 63;
        wlds[k * 65 + n] = (k < KDIM) ? W[(size_t)k * 64 + n] : 0.0f;
    }
    __syncthreads();

    const int gwave  = (blockIdx.x * blockDim.x + threadIdx.x) >> 5;
    const int lane   = threadIdx.x & 31;
    const int tile   = gwave;
    const int ntiles = (nrows + 15) >> 4;
    if (tile >= ntiles) return;              // wave-uniform exit: EXEC all-1s for WMMA

    const int m     = lane & 15;             // A row within tile / B,C,D column
    const int khalf = lane >> 4;             // 0: K pair {0,1}, 1: K pair {2,3}
    const int row   = min(tile * 16 + m, nrows - 1);   // clamp, no divergence

    v8f acc[4] = {};                         // 4 n-tiles of 16 cols each

#pragma unroll
    for (int ks = 0; ks < KSTEPS; ++ks) {
        const int k0 = ks * 4 + khalf * 2;
        v2f a;
        if (KDIM == 64) {
            a = *(const v2f*)(X + (size_t)row * 64 + k0);
        } else {
            const bool ok = (k0 + 1 < KDIM);
            a[0] = X[(size_t)row * KDIM + k0];
            // clamped (always-valid) address, value selected to 0 -> v_cndmask, no branch
            const float a1 = X[(size_t)row * KDIM + (ok ? k0 + 1 : k0)];
            a[1] = ok ? a1 : 0.0f;
        }
        if (RELU) { a[0] = fmaxf(a[0], 0.0f); a[1] = fmaxf(a[1], 0.0f); }

#pragma unroll
        for (int nt = 0; nt < 4; ++nt) {
            v2f b;
            b[0] = wlds[k0 * 65 + nt * 16 + m];
            b[1] = wlds[(k0 + 1) * 65 + nt * 16 + m];
            acc[nt] = __builtin_amdgcn_wmma_f32_16x16x4_f32(
                false, a, false, b, (short)0, acc[nt], false, false);
        }
    }

    // C/D layout: VGPR r -> M = r + 8*khalf, N = lane & 15
    const int col   = lane & 15;
    const int rbase = khalf * 8;
    if (tile * 16 + 16 <= nrows) {
        // full tile (always taken when nrows % 16 == 0): unguarded store clause
#pragma unroll
        for (int nt = 0; nt < 4; ++nt) {
#pragma unroll
            for (int r = 0; r < 8; ++r) {
                const int orow = tile * 16 + rbase + r;
                H[(size_t)orow * 64 + nt * 16 + col] = acc[nt][r];
            }
        }
    } else {
#pragma unroll
        for (int nt = 0; nt < 4; ++nt) {
#pragma unroll
            for (int r = 0; r < 8; ++r) {
                const int orow = tile * 16 + rbase + r;
                if (orow < nrows) H[(size_t)orow * 64 + nt * 16 + col] = acc[nt][r];
            }
        }
    }
}

// ---------------- aggregation init: tmp[i][f] = b[f] + dinv[i]^2 * h[i][f] ----------------
__global__ void k_agg_init(const float* __restrict__ h, const float* __restrict__ dinv,
                           const float* __restrict__ bias, float* __restrict__ tmp, int n) {
    const int tid = blockIdx.x * blockDim.x + threadIdx.x;
    const int i = tid >> 6;
    if (i >= n) return;
    const int f = tid & 63;
    const float di = dinv[i];
    tmp[(size_t)i * 64 + f] = bias[f] + di * di * h[(size_t)i * 64 + f];
}

// ---------------- edge scatter: tmp[dst] += norm * h[src]; 16 threads/edge, float4 each ----
__global__ void k_scatter(const float* __restrict__ h, const int* __restrict__ src,
                          const int* __restrict__ dst, const float* __restrict__ nrm,
                          float* __restrict__ tmp, int E) {
    const int tid = blockIdx.x * blockDim.x + threadIdx.x;
    const int e = tid >> 4;
    if (e >= E) return;
    const int f4 = (tid & 15) << 2;
    const int s = src[e];
    const int d = dst[e];
    const float w = nrm[e];
    const float4 v = *(const float4*)(h + (size_t)s * 64 + f4);
    float* o = tmp + (size_t)d * 64 + f4;
    atomicAdd(o + 0, w * v.x);
    atomicAdd(o + 1, w * v.y);
    atomicAdd(o + 2, w * v.z);
    atomicAdd(o + 3, w * v.w);
}

// ---------------- pooling: out[batch[i]][f] += relu(tmp[i][f]) ----------------
__global__ void k_pool(const float* __restrict__ tmp, const int* __restrict__ batch,
                       float* __restrict__ out, int n) {
    const int tid = blockIdx.x * blockDim.x + threadIdx.x;
    const int i = tid >> 6;
    if (i >= n) return;
    const int f = tid & 63;
    const float v = fmaxf(tmp[(size_t)i * 64 + f], 0.0f);
    atomicAdd(out + batch[i] * 64 + f, v);
}

__global__ void k_final(float* __restrict__ out, const float* __restrict__ cnt, int n) {
    int i = blockIdx.x * blockDim.x + threadIdx.x;
    if (i < n) out[i] /= fmaxf(cnt[i >> 6], 1.0f);
}

// ---------------- host ----------------
static inline unsigned cdiv(long long a, long long b) { return (unsigned)((a + b - 1) / b); }

extern "C" void kernel_launch(void* const* d_in, const int* in_sizes, int n_in,
                              void* d_out, int out_size, void* d_ws, size_t ws_size,
                              hipStream_t stream) {
    const float* x   = (const float*)d_in[0];
    const int*   ei  = (const int*)d_in[1];
    const int*   bat = (const int*)d_in[2];
    const float* W1  = (const float*)d_in[3];
    const float* b1  = (const float*)d_in[4];
    const float* W2  = (const float*)d_in[5];
    const float* b2  = (const float*)d_in[6];
    const float* W3  = (const float*)d_in[7];
    const float* b3  = (const float*)d_in[8];
    const float* W4  = (const float*)d_in[9];
    const float* b4  = (const float*)d_in[10];
    float* out = (float*)d_out;

    const int N = in_sizes[0] / 15;
    const int E = in_sizes[1] / 2;
    const int* src = ei;
    const int* dst = ei + E;

    float* ws   = (float*)d_ws;
    float* h    = ws;                         // N*64
    float* tmp  = h + (size_t)N * 64;         // N*64
    float* dinv = tmp + (size_t)N * 64;       // N   (also used as deg)
    float* nrm  = dinv + N;                   // E
    float* cnt  = nrm + E;                    // 256

    const int T = 256;

    // normalization: deg (self-loop baked in as 1.0), dinv, per-edge norm
    k_fill<<<cdiv(N, T), T, 0, stream>>>(dinv, 1.0f, N);
    k_deg<<<cdiv(E, T), T, 0, stream>>>(dst, dinv, E);
    k_rsqrt<<<cdiv(N, T), T, 0, stream>>>(dinv, N);
    k_norm<<<cdiv(E, T), T, 0, stream>>>(src, dst, dinv, nrm, E);

    // pooling accumulators
    k_fill<<<cdiv(256 * 64, T), T, 0, stream>>>(out, 0.0f, 256 * 64);
    k_fill<<<1, 256, 0, stream>>>(cnt, 0.0f, 256);
    k_cnt<<<cdiv(N, T), T, 0, stream>>>(bat, cnt, N);

    const unsigned gemmBlocks = cdiv((N + 15) / 16, T / 32);  // 1 wave per 16-row tile
    const unsigned nodeF      = cdiv((long long)N * 64, T);
    const unsigned edgeF      = cdiv((long long)E * 16, T);

    // layer 1 (K=15, no input relu)
    k_gemm<15, false><<<gemmBlocks, T, 0, stream>>>(x, W1, h, N);
    k_agg_init<<<nodeF, T, 0, stream>>>(h, dinv, b1, tmp, N);
    k_scatter<<<edgeF, T, 0, stream>>>(h, src, dst, nrm, tmp, E);
    // layer 2
    k_gemm<64, true><<<gemmBlocks, T, 0, stream>>>(tmp, W2, h, N);
    k_agg_init<<<nodeF, T, 0, stream>>>(h, dinv, b2, tmp, N);
    k_scatter<<<edgeF, T, 0, stream>>>(h, src, dst, nrm, tmp, E);
    // layer 3
    k_gemm<64, true><<<gemmBlocks, T, 0, stream>>>(tmp, W3, h, N);
    k_agg_init<<<nodeF, T, 0, stream>>>(h, dinv, b3, tmp, N);
    k_scatter<<<edgeF, T, 0, stream>>>(h, src, dst, nrm, tmp, E);
    // layer 4
    k_gemm<64, true><<<gemmBlocks, T, 0, stream>>>(tmp, W4, h, N);
    k_agg_init<<<nodeF, T, 0, stream>>>(h, dinv, b4, tmp, N);
    k_scatter<<<edgeF, T, 0, stream>>>(h, src, dst, nrm, tmp, E);

    // global mean pool (relu fused into the read)
    k_pool<<<nodeF, T, 0, stream>>>(tmp, bat, out, N);
    k_final<<<cdiv(256 * 64, T), T, 0, stream>>>(out, cnt, 256 * 64);
}